// Net_86517821216404
// MI455X (gfx1250) — compile-verified
//
#include <hip/hip_runtime.h>
#include <hip/hip_bf16.h>

typedef __attribute__((ext_vector_type(2))) float v2f;
typedef __attribute__((ext_vector_type(8))) float v8f;

// ---------------------------------------------------------------------------
// 0) zero workspace
// ---------------------------------------------------------------------------
__global__ void zero_kernel(float* __restrict__ p, int n) {
    int i = blockIdx.x * blockDim.x + threadIdx.x;
    if (i < n) p[i] = 0.0f;
}

// ---------------------------------------------------------------------------
// 1) degree count: deg[dst] += 1 over the 2E symmetrized edges
//    edge_index flat layout: src = ei[e], dst = ei[NE + e]
// ---------------------------------------------------------------------------
__global__ void deg_kernel(const int* __restrict__ ei, int NE, float* __restrict__ deg) {
    int e = blockIdx.x * blockDim.x + threadIdx.x;
    if (e < NE) atomicAdd(&deg[ei[NE + e]], 1.0f);
}

// ---------------------------------------------------------------------------
// 2) per-edge GCN norm = dinv[src]*dinv[dst]
// ---------------------------------------------------------------------------
__global__ void norm_kernel(const int* __restrict__ ei, int NE,
                            const float* __restrict__ deg, float* __restrict__ nrm) {
    int e = blockIdx.x * blockDim.x + threadIdx.x;
    if (e >= NE) return;
    int s = ei[e], d = ei[NE + e];
    float ds = deg[s], dd = deg[d];
    float is = (ds > 0.0f) ? rsqrtf(fmaxf(ds, 1.0f)) : 0.0f;
    float id = (dd > 0.0f) ? rsqrtf(fmaxf(dd, 1.0f)) : 0.0f;
    nrm[e] = is * id;
}

// ---------------------------------------------------------------------------
// gene-sorted 4-peak scatter combine (usually 1 atomic instead of 4)
// ---------------------------------------------------------------------------
__device__ __forceinline__ void scatter4(float* __restrict__ accb, float4 z,
                                         int g0, int g1, int g2, int g3) {
    if (g0 == g3) {
        atomicAdd(accb + g0, (z.x + z.y) + (z.z + z.w));
    } else {
        if (g0 == g1) atomicAdd(accb + g0, z.x + z.y);
        else { atomicAdd(accb + g0, z.x); atomicAdd(accb + g1, z.y); }
        if (g2 == g3) atomicAdd(accb + g2, z.z + z.w);
        else { atomicAdd(accb + g2, z.z); atomicAdd(accb + g3, z.w); }
    }
}

// ---------------------------------------------------------------------------
// 3) dominant streaming kernel: z[b,p] = sum_tf x[b,tf,p]*w[tf,p],
//    fused segment-scatter: acc[b, gene_ids[p]] += z[b,p]
//    448 MB of x streamed once (HBM roofline ~19us @ 23.3 TB/s).
//    Two batches per thread: each w_sub load (L2-resident) feeds 2 FMAs.
// ---------------------------------------------------------------------------
__global__ void reduce_scatter_kernel(const float* __restrict__ x,
                                      const float* __restrict__ w,
                                      const int*   __restrict__ gid,
                                      float*       __restrict__ acc,
                                      int T, int TF, int G, int B) {
    const int b0 = blockIdx.y * 2;
    const int b1 = b0 + 1;
    const int p0 = (blockIdx.x * blockDim.x + threadIdx.x) * 4;
    if (p0 >= T) return;

    const float* __restrict__ xb0 = x + (size_t)b0 * TF * T;
    const float* __restrict__ xb1 = x + (size_t)((b1 < B) ? b1 : b0) * TF * T;

    if (p0 + 4 <= T) {
        float4 z0 = make_float4(0.f, 0.f, 0.f, 0.f);
        float4 z1 = make_float4(0.f, 0.f, 0.f, 0.f);
        for (int tf = 0; tf < TF; ++tf) {
            const size_t off = (size_t)tf * T + p0;
            const float4 wv  = *(const float4*)(w   + off);
            const float4 xv0 = *(const float4*)(xb0 + off);
            const float4 xv1 = *(const float4*)(xb1 + off);
            if (tf + 8 < TF) {  // CDNA5 speculative prefetch down the strided tf walk
                __builtin_prefetch(xb0 + off + (size_t)8 * T, 0, 1);
                __builtin_prefetch(xb1 + off + (size_t)8 * T, 0, 1);
            }
            z0.x = fmaf(xv0.x, wv.x, z0.x);
            z0.y = fmaf(xv0.y, wv.y, z0.y);
            z0.z = fmaf(xv0.z, wv.z, z0.z);
            z0.w = fmaf(xv0.w, wv.w, z0.w);
            z1.x = fmaf(xv1.x, wv.x, z1.x);
            z1.y = fmaf(xv1.y, wv.y, z1.y);
            z1.z = fmaf(xv1.z, wv.z, z1.z);
            z1.w = fmaf(xv1.w, wv.w, z1.w);
        }
        const int g0 = gid[p0], g1 = gid[p0 + 1], g2 = gid[p0 + 2], g3 = gid[p0 + 3];
        scatter4(acc + (size_t)b0 * G, z0, g0, g1, g2, g3);
        if (b1 < B) scatter4(acc + (size_t)b1 * G, z1, g0, g1, g2, g3);
    } else {  // ragged tail (not hit for T=54736, kept for generality)
        for (int p = p0; p < T; ++p) {
            float s0 = 0.f, s1 = 0.f;
            for (int tf = 0; tf < TF; ++tf) {
                const size_t off = (size_t)tf * T + p;
                const float wv = w[off];
                s0 = fmaf(xb0[off], wv, s0);
                s1 = fmaf(xb1[off], wv, s1);
            }
            atomicAdd(acc + (size_t)b0 * G + gid[p], s0);
            if (b1 < B) atomicAdd(acc + (size_t)b1 * G + gid[p], s1);
        }
    }
}

// ---------------------------------------------------------------------------
// 4) x_cat = relu(acc + b_sub)  -> d_out region 0  [B,G]
// ---------------------------------------------------------------------------
__global__ void xcat_kernel(const float* __restrict__ acc, const float* __restrict__ b_sub,
                            float* __restrict__ xcat, int BG, int G) {
    int i = blockIdx.x * blockDim.x + threadIdx.x;
    if (i >= BG) return;
    float v = acc[i] + b_sub[i % G];
    xcat[i] = v > 0.0f ? v : 0.0f;
}

// ---------------------------------------------------------------------------
// 5) GCN scatter (w_conv factored out): S[b,dst] += x_cat[b,src]*norm[e]
// ---------------------------------------------------------------------------
__global__ void gcn_scatter_kernel(const int* __restrict__ ei, int NE,
                                   const float* __restrict__ nrm,
                                   const float* __restrict__ xcat,
                                   float* __restrict__ S, int G) {
    int e = blockIdx.x * blockDim.x + threadIdx.x;
    if (e >= NE) return;
    const int b = blockIdx.y;
    const int s = ei[e], d = ei[NE + e];
    atomicAdd(&S[(size_t)b * G + d], xcat[(size_t)b * G + s] * nrm[e]);
}

// ---------------------------------------------------------------------------
// 6) g[b, gg*2+f] = relu(w_conv[f]*S[b,gg] + b_conv[f]) -> d_out region 1 [B,2G]
// ---------------------------------------------------------------------------
__global__ void g_kernel(const float* __restrict__ S, const float* __restrict__ w_conv,
                         const float* __restrict__ b_conv, float* __restrict__ gmat,
                         int BG, int G) {
    int i = blockIdx.x * blockDim.x + threadIdx.x;
    if (i >= BG) return;
    const int b = i / G, gg = i % G;
    const float s = S[i];
    float f0 = fmaf(w_conv[0], s, b_conv[0]);
    float f1 = fmaf(w_conv[1], s, b_conv[1]);
    float* o = gmat + (size_t)b * (2 * G) + (size_t)gg * 2;
    o[0] = f0 > 0.0f ? f0 : 0.0f;
    o[1] = f1 > 0.0f ? f1 : 0.0f;
}

// ---------------------------------------------------------------------------
// 7) out = g @ w_out + b_out  via V_WMMA_F32_16X16X4_F32
//    [32 x K=2G] * [K x 3] ; 2 waves, each owns a 16-row M tile, K/4 WMMAs.
//    w_out panel (48 KB) staged in LDS; B operand fed via ds_load.
//    Branch-free inner loop: pad columns handled with clamped index + mask
//    (loop-invariant), so EXEC stays all-1s as WMMA requires.
//    A layout (ISA 16x4 f32): m = lane%16, k = (lane/16)*2 + vgpr.
//    B layout (mirror):       n = lane%16, k = (lane/16)*2 + vgpr.
//    D layout (16x16 f32 C/D): row = vgpr + 8*(lane/16), col = lane%16.
// ---------------------------------------------------------------------------
__global__ void out_gemm_wmma_kernel(const float* __restrict__ g,
                                     const float* __restrict__ w_out,
                                     const float* __restrict__ b_out,
                                     float* __restrict__ out,
                                     int K, int N) {
    extern __shared__ float lds_w[];          // K*N floats (12000 = 48 KB)
    const int tid = threadIdx.x;
    for (int i = tid; i < K * N; i += blockDim.x) lds_w[i] = w_out[i];
    __syncthreads();

    const int lane = tid & 31;
    const int wave = tid >> 5;               // 0..1 -> M tile
    const int half = lane >> 4;              // 0/1
    const int mrow = (lane & 15) + wave * 16;
    const int ncol = lane & 15;
    const int nc   = (ncol < N) ? ncol : 0;          // loop-invariant clamp
    const float msk = (ncol < N) ? 1.0f : 0.0f;      // loop-invariant mask

    const float* __restrict__ grow = g + (size_t)mrow * K;

    v8f c = {};
    for (int k = 0; k < K; k += 4) {
        const int ke = k + half * 2;
        v2f a, bm;
        a.x  = grow[ke];
        a.y  = grow[ke + 1];
        bm.x = lds_w[ke * N + nc] * msk;
        bm.y = lds_w[(ke + 1) * N + nc] * msk;
        c = __builtin_amdgcn_wmma_f32_16x16x4_f32(
                /*neg_a=*/false, a, /*neg_b=*/false, bm,
                /*c_mod=*/(short)0, c, /*reuse_a=*/false, /*reuse_b=*/false);
    }
    if (ncol < N) {
        const float bias = b_out[ncol];
        #pragma unroll
        for (int r = 0; r < 8; ++r) {
            const int m = r + 8 * half + wave * 16;
            out[(size_t)m * N + ncol] = c[r] + bias;
        }
    }
}

// ---------------------------------------------------------------------------
extern "C" void kernel_launch(void* const* d_in, const int* in_sizes, int n_in,
                              void* d_out, int out_size, void* d_ws, size_t ws_size,
                              hipStream_t stream) {
    const float* x      = (const float*)d_in[0];
    const int*   gid    = (const int*)  d_in[1];
    const float* w_sub  = (const float*)d_in[2];
    const float* b_sub  = (const float*)d_in[3];
    const int*   ei     = (const int*)  d_in[4];
    const float* w_conv = (const float*)d_in[5];
    const float* b_conv = (const float*)d_in[6];
    const float* w_out  = (const float*)d_in[7];
    const float* b_out  = (const float*)d_in[8];
    float*       out    = (float*)d_out;

    const int T  = in_sizes[1];               // 54736 peak columns
    const int TF = in_sizes[2] / T;           // 64
    const int B  = in_sizes[0] / in_sizes[2]; // 32
    const int G  = in_sizes[3];               // 2000
    const int NE = in_sizes[4] / 2;           // 64000 symmetrized edges
    const int BG = B * G;

    // workspace layout (floats)
    float* acc  = (float*)d_ws;            // [B*G] pre-activation x_cat accum
    float* S    = acc + (size_t)BG;        // [B*G] GCN scalar scatter accum
    float* deg  = S   + (size_t)BG;        // [G]
    float* nrm  = deg + (size_t)G;         // [NE]
    const int Nz = 2 * BG + G + NE;

    // d_out regions: x_cat [B,G] | g [B,2G] | out [B,3]
    float* xcat_out  = out;
    float* g_out     = out + (size_t)BG;
    float* final_out = out + (size_t)BG * 3;

    zero_kernel<<<(Nz + 255) / 256, 256, 0, stream>>>((float*)d_ws, Nz);
    deg_kernel<<<(NE + 255) / 256, 256, 0, stream>>>(ei, NE, deg);
    norm_kernel<<<(NE + 255) / 256, 256, 0, stream>>>(ei, NE, deg, nrm);

    dim3 gr((T / 4 + 255) / 256, (B + 1) / 2);
    reduce_scatter_kernel<<<gr, 256, 0, stream>>>(x, w_sub, gid, acc, T, TF, G, B);

    xcat_kernel<<<(BG + 255) / 256, 256, 0, stream>>>(acc, b_sub, xcat_out, BG, G);

    dim3 ge((NE + 255) / 256, B);
    gcn_scatter_kernel<<<ge, 256, 0, stream>>>(ei, NE, nrm, xcat_out, S, G);

    g_kernel<<<(BG + 255) / 256, 256, 0, stream>>>(S, w_conv, b_conv, g_out, BG, G);

    const int K = 2 * G, N = 3;
    out_gemm_wmma_kernel<<<1, 64, (size_t)K * N * sizeof(float), stream>>>(
        g_out, w_out, b_out, final_out, K, N);
}